// O2BWeightedDeltaNetLayer_69982197121468
// MI455X (gfx1250) — compile-verified
//
#include <hip/hip_runtime.h>
#include <hip/hip_bf16.h>

typedef float v2f __attribute__((ext_vector_type(2)));
typedef float v8f __attribute__((ext_vector_type(8)));

#define WMMA_F32(a, b, c) \
  __builtin_amdgcn_wmma_f32_16x16x4_f32(false, (a), false, (b), (short)0, (c), false, false)

constexpr int Bb    = 4;
constexpr int Lseq  = 2048;
constexpr int Dm    = 1024;
constexpr int H     = 16;
constexpr int Dh    = 64;
constexpr int CK    = 4;      // conv kernel width
constexpr int C     = 64;     // chunk
constexpr int Nc    = Lseq / C;
constexpr int Mrows = Bb * Lseq;   // 8192

// ---------------------------------------------------------------------------
// Generic Y = X @ W^T + bias GEMM.  X:(M,K) row-major, W:(N,K) row-major.
// 64x64 tile per 512-thread block (16 waves, each one 16x16 WMMA tile).
// ---------------------------------------------------------------------------
__global__ __launch_bounds__(512)
void gemm_xwt_kernel(const float* __restrict__ X, const float* __restrict__ W,
                     const float* __restrict__ bias, float* __restrict__ Y,
                     int M, int N, int K) {
  __shared__ float Xs[64][65];
  __shared__ float Ws[64][65];
  const int tid  = threadIdx.x;
  const int wave = tid >> 5, lane = tid & 31;
  const int h = lane >> 4, l = lane & 15;
  const int wm = (wave >> 2) * 16, wn = (wave & 3) * 16;
  const int m0 = blockIdx.y * 64, n0 = blockIdx.x * 64;
  const int lr = tid >> 3;          // 0..63 (tile row to load)
  const int lc = (tid & 7) * 8;     // col start (8 floats per thread)

  v8f acc = {};
  for (int k0 = 0; k0 < K; k0 += 64) {
    float4 xa = *(const float4*)(X + (size_t)(m0 + lr) * K + k0 + lc);
    float4 xb = *(const float4*)(X + (size_t)(m0 + lr) * K + k0 + lc + 4);
    float4 wa = *(const float4*)(W + (size_t)(n0 + lr) * K + k0 + lc);
    float4 wb = *(const float4*)(W + (size_t)(n0 + lr) * K + k0 + lc + 4);
    __syncthreads();
    Xs[lr][lc + 0] = xa.x; Xs[lr][lc + 1] = xa.y; Xs[lr][lc + 2] = xa.z; Xs[lr][lc + 3] = xa.w;
    Xs[lr][lc + 4] = xb.x; Xs[lr][lc + 5] = xb.y; Xs[lr][lc + 6] = xb.z; Xs[lr][lc + 7] = xb.w;
    Ws[lr][lc + 0] = wa.x; Ws[lr][lc + 1] = wa.y; Ws[lr][lc + 2] = wa.z; Ws[lr][lc + 3] = wa.w;
    Ws[lr][lc + 4] = wb.x; Ws[lr][lc + 5] = wb.y; Ws[lr][lc + 6] = wb.z; Ws[lr][lc + 7] = wb.w;
    __syncthreads();
#pragma unroll
    for (int kk = 0; kk < 64; kk += 4) {
      const int ka = kk + 2 * h;          // lanes 16-31 carry K = 2,3 of the step
      v2f av, bv;
      av.x = Xs[wm + l][ka];  av.y = Xs[wm + l][ka + 1];
      bv.x = Ws[wn + l][ka];  bv.y = Ws[wn + l][ka + 1];
      acc = WMMA_F32(av, bv, acc);
    }
  }
#pragma unroll
  for (int i = 0; i < 8; i++) {
    const int m = m0 + wm + 8 * h + i;    // vgpr i -> M = i + 8*half
    const int n = n0 + wn + l;
    Y[(size_t)m * N + n] = acc[i] + bias[n];
  }
}

// ---------------------------------------------------------------------------
// beta = sigmoid(x @ Wbeta^T + bbeta), shape (Mrows, H)
// ---------------------------------------------------------------------------
__global__ void beta_kernel(const float* __restrict__ X, const float* __restrict__ Wb,
                            const float* __restrict__ bb, float* __restrict__ beta) {
  const int t = blockIdx.x * blockDim.x + threadIdx.x;
  if (t >= Mrows * H) return;
  const int m = t >> 4, h = t & 15;
  const float* xr = X + (size_t)m * Dm;
  const float* wr = Wb + (size_t)h * Dm;
  float s = bb[h];
  for (int k = 0; k < Dm; k += 4) {
    s += xr[k] * wr[k] + xr[k + 1] * wr[k + 1] + xr[k + 2] * wr[k + 2] + xr[k + 3] * wr[k + 3];
  }
  beta[t] = 1.f / (1.f + expf(-s));
}

// ---------------------------------------------------------------------------
// Causal depthwise conv along L:  y[b,l,c] = sum_j x[b, l-3+j, c] * w[c,j]
// ---------------------------------------------------------------------------
__global__ void conv_kernel(const float* __restrict__ P, const float* __restrict__ Wc,
                            float* __restrict__ Yc) {
  const int e = blockIdx.x * blockDim.x + threadIdx.x;
  if (e >= Mrows * Dm) return;
  const int c = e & (Dm - 1);
  const int m = e >> 10;          // Dm = 1024
  const int l = m & (Lseq - 1);
  float s = 0.f;
#pragma unroll
  for (int j = 0; j < CK; j++) {
    const int ls = l + j - (CK - 1);
    if (ls >= 0) s += P[(size_t)(m + j - (CK - 1)) * Dm + c] * Wc[c * CK + j];
  }
  Yc[e] = s;
}

// ---------------------------------------------------------------------------
// Per-(b,l,h): normalize k,q; v = beta*silu(v); b = beta*k.
// One wave per row; outputs in [bh][l][Dh] layout.
// ---------------------------------------------------------------------------
__global__ __launch_bounds__(256)
void prep_kernel(const float* __restrict__ kc, const float* __restrict__ qc,
                 const float* __restrict__ vc, const float* __restrict__ beta,
                 float* __restrict__ Kp, float* __restrict__ Qp,
                 float* __restrict__ Vp, float* __restrict__ Bp) {
  const int row  = blockIdx.x * 8 + (threadIdx.x >> 5);   // over (b*L+l)*H + h
  if (row >= Mrows * H) return;
  const int lane = threadIdx.x & 31;
  const int h  = row & (H - 1);
  const int ml = row >> 4;                 // b*L + l
  const int b  = ml / Lseq, l = ml & (Lseq - 1);
  const size_t src = (size_t)ml * Dm + h * Dh;
  float k0v = kc[src + lane], k1v = kc[src + lane + 32];
  float q0v = qc[src + lane], q1v = qc[src + lane + 32];
  float v0v = vc[src + lane], v1v = vc[src + lane + 32];
  float ks = k0v * k0v + k1v * k1v;
  float qs = q0v * q0v + q1v * q1v;
#pragma unroll
  for (int off = 16; off > 0; off >>= 1) {
    ks += __shfl_xor(ks, off);
    qs += __shfl_xor(qs, off);
  }
  const float kin = 1.f / (sqrtf(ks) + 1e-6f);
  const float qin = 1.f / (sqrtf(qs) + 1e-6f);
  const float bet = beta[(size_t)ml * H + h];
  const float s0 = v0v / (1.f + expf(-v0v));
  const float s1 = v1v / (1.f + expf(-v1v));
  const int bh = b * H + h;
  const size_t dst = ((size_t)bh * Lseq + l) * Dh;
  const float kn0 = k0v * kin, kn1 = k1v * kin;
  Kp[dst + lane] = kn0;             Kp[dst + lane + 32] = kn1;
  Qp[dst + lane] = q0v * qin;       Qp[dst + lane + 32] = q1v * qin;
  Vp[dst + lane] = bet * s0;        Vp[dst + lane + 32] = bet * s1;
  Bp[dst + lane] = bet * kn0;       Bp[dst + lane + 32] = bet * kn1;
}

// ---------------------------------------------------------------------------
// Delta-rule scan.  One 512-thread block per (b,h).  All 64x64 matmuls via
// WMMA fully in LDS (row stride 65 to dodge bank conflicts).
// ---------------------------------------------------------------------------
constexpr int ST = 65;
constexpr int SZ = 64 * ST;

__device__ __forceinline__ float lds_at(const float* M, bool T, int r, int c) {
  return T ? M[c * ST + r] : M[r * ST + c];
}

// C(64x64) = opA(A) @ opB(B); optional per-k scale on A (ksc), per-row scale
// on the result (rsc), accumulate-into-LDS.  All 512 threads participate.
__device__ __forceinline__ void mm64(const float* A, bool TA, const float* B, bool TB,
                                     float* Cb, const float* ksc, const float* rsc,
                                     bool accum, int tid) {
  const int wave = tid >> 5, lane = tid & 31;
  const int h = lane >> 4, l = lane & 15;
  const int wm = (wave >> 2) * 16, wn = (wave & 3) * 16;
  v8f acc = {};
#pragma unroll
  for (int kk = 0; kk < 64; kk += 4) {
    const int ka = kk + 2 * h;
    v2f av, bv;
    av.x = lds_at(A, TA, wm + l, ka);
    av.y = lds_at(A, TA, wm + l, ka + 1);
    bv.x = TB ? B[(wn + l) * ST + ka]     : B[ka * ST + (wn + l)];
    bv.y = TB ? B[(wn + l) * ST + ka + 1] : B[(ka + 1) * ST + (wn + l)];
    if (ksc) { av.x *= ksc[ka]; av.y *= ksc[ka + 1]; }
    acc = WMMA_F32(av, bv, acc);
  }
#pragma unroll
  for (int i = 0; i < 8; i++) {
    const int m = wm + 8 * h + i, n = wn + l;
    float v = acc[i];
    if (rsc) v *= rsc[m];
    if (accum) Cb[m * ST + n] += v; else Cb[m * ST + n] = v;
  }
}

__global__ __launch_bounds__(512)
void delta_kernel(const float* __restrict__ Kp, const float* __restrict__ Qp,
                  const float* __restrict__ Vp, const float* __restrict__ Bp,
                  float* __restrict__ Og) {
  extern __shared__ float sm[];
  float* Kc = sm + 0 * SZ;
  float* Qc = sm + 1 * SZ;
  float* BO = sm + 2 * SZ;   // Bc chunk, reused as O accumulator
  float* Wt = sm + 3 * SZ;
  float* Wa = sm + 4 * SZ;
  float* T0 = sm + 5 * SZ;   // S = tril(B K^T), later QK
  float* T1 = sm + 6 * SZ;   // RHS -> U (in-place solve)
  float* T2 = sm + 7 * SZ;   // B@Wt, then E, then K^T(U*w)
  float* T3 = sm + 8 * SZ;   // K^T U
  __shared__ float cs_s[64], wi_s[64], a_s[64], c_s[64];

  const int tid = threadIdx.x;
  const int bh  = blockIdx.x;
  const size_t base = (size_t)bh * Lseq * Dh;

  for (int e = tid; e < SZ; e += 512) { Wt[e] = 0.f; Wa[e] = 0.f; }
  __syncthreads();

  for (int i = 0; i < Nc; i++) {
    const float ts = (float)(i * C);
    if (tid < 64) {
      const float r1  = (float)(tid + 1);
      const float idx = ts + r1;
      const float cs  = r1 * ts + r1 * (r1 + 1.f) * 0.5f;
      cs_s[tid] = cs;
      const float sumidx = 64.f * ts + 2080.f;              // sum_{t=1..64}(ts+t)
      const float denom  = (1.f + ts + 64.f) * (ts + 64.f) * 0.5f;
      wi_s[tid] = (idx + sumidx - cs) / denom;
      const float tri0 = (1.f + ts) * ts * 0.5f;
      const float a = tri0 / (tri0 + cs);
      a_s[tid] = a;  c_s[tid] = 1.f - a;
    }
    const size_t cb = base + (size_t)i * C * Dh;
    for (int e = tid; e < C * Dh; e += 512) {
      const int r = e >> 6, d = e & 63;
      Kc[r * ST + d] = Kp[cb + e];
      Qc[r * ST + d] = Qp[cb + e];
      BO[r * ST + d] = Bp[cb + e];
    }
    __syncthreads();

    mm64(BO, false, Kc, true,  T0, nullptr, nullptr, false, tid);   // S  = B K^T
    mm64(BO, false, Wt, false, T2, nullptr, nullptr, false, tid);   // P  = B Wt
    __syncthreads();

    for (int e = tid; e < C * Dh; e += 512) {                       // RHS = V - P
      const int r = e >> 6, d = e & 63;
      T1[r * ST + d] = Vp[cb + e] - T2[r * ST + d];
    }
    __syncthreads();

    // Solve (I + tril(S,-1)) U = RHS; column-parallel, each column local.
    if (tid < 64) {
      const int cc = tid;
      for (int r = 1; r < C; r++) {
        float acc = T1[r * ST + cc];
        for (int j = 0; j < r; j++) acc -= T0[r * ST + j] * T1[j * ST + cc];
        T1[r * ST + cc] = acc;
      }
    }
    __syncthreads();

    mm64(Qc, false, Kc, true, T0, nullptr, nullptr, false, tid);    // QK = Q K^T
    __syncthreads();

    {                                                               // E = T_mat o QK
      const float tri0 = (1.f + ts) * ts * 0.5f;
      for (int e = tid; e < C * C; e += 512) {
        const int r = e >> 6, j = e & 63;
        const float csr  = cs_s[r];
        const float jidx = cs_s[j] - (ts + (float)(j + 1));
        const float tm = (csr >= jidx) ? (csr - jidx) / (tri0 + csr) : 0.f;
        T2[r * ST + j] = tm * T0[r * ST + j];
      }
    }
    __syncthreads();

    mm64(Qc, false, Wa, false, BO, nullptr, a_s, false, tid);       // O  = a*(Q Wa)
    __syncthreads();
    mm64(Qc, false, Wt, false, BO, nullptr, c_s, true, tid);        // O += c*(Q Wt)
    __syncthreads();
    mm64(T2, false, T1, false, BO, nullptr, nullptr, true, tid);    // O += E U
    __syncthreads();

    for (int e = tid; e < C * Dh; e += 512) {
      const int r = e >> 6, d = e & 63;
      Og[cb + e] = BO[r * ST + d];
    }

    mm64(Kc, true, T1, false, T3, nullptr, nullptr, false, tid);    // K^T U
    mm64(Kc, true, T1, false, T2, wi_s,   nullptr, false, tid);     // K^T (U * w_i)
    __syncthreads();

    {
      const float coef1 = ts / (ts + 64.f) * ((1.f + ts) / (1.f + ts + 64.f));
      const float coef2 = 64.f / (ts + 64.f) * ((ts * 2.f + 1.f + 64.f) / (1.f + ts + 64.f));
      for (int e = tid; e < C * Dh; e += 512) {
        const int r = e >> 6, d = e & 63, o = r * ST + d;
        const float wt = Wt[o], wa = Wa[o];
        Wa[o] = coef1 * wa + coef2 * wt + T2[o];
        Wt[o] = wt + T3[o];
      }
    }
    __syncthreads();
  }
}

// ---------------------------------------------------------------------------
// RMS norm over Dh; scatter from [bh][l][Dh] to (B,L,Dm) layout.
// ---------------------------------------------------------------------------
__global__ __launch_bounds__(256)
void rms_kernel(const float* __restrict__ Og, const float* __restrict__ rmsw,
                float* __restrict__ On) {
  const int row = blockIdx.x * 8 + (threadIdx.x >> 5);   // bh*L + l
  if (row >= Bb * H * Lseq) return;
  const int lane = threadIdx.x & 31;
  const int bh = row / Lseq, l = row & (Lseq - 1);
  const int b = bh >> 4, h = bh & 15;
  const size_t src = (size_t)row * Dh;
  const float o0 = Og[src + lane], o1 = Og[src + lane + 32];
  float ss = o0 * o0 + o1 * o1;
#pragma unroll
  for (int off = 16; off > 0; off >>= 1) ss += __shfl_xor(ss, off);
  const float inv = rsqrtf(ss * (1.f / 64.f) + 1e-6f);
  const size_t dst = ((size_t)(b * Lseq + l)) * Dm + h * Dh;
  On[dst + lane]      = o0 * inv * rmsw[lane];
  On[dst + lane + 32] = o1 * inv * rmsw[lane + 32];
}

// ---------------------------------------------------------------------------
extern "C" void kernel_launch(void* const* d_in, const int* in_sizes, int n_in,
                              void* d_out, int out_size, void* d_ws, size_t ws_size,
                              hipStream_t stream) {
  (void)in_sizes; (void)n_in; (void)out_size; (void)ws_size;
  const float* x     = (const float*)d_in[0];
  const float* Wk    = (const float*)d_in[1];
  const float* bk    = (const float*)d_in[2];
  const float* Wq    = (const float*)d_in[3];
  const float* bq    = (const float*)d_in[4];
  const float* Wv    = (const float*)d_in[5];
  const float* bv    = (const float*)d_in[6];
  const float* Wbeta = (const float*)d_in[7];
  const float* bbeta = (const float*)d_in[8];
  const float* convk = (const float*)d_in[9];
  const float* convq = (const float*)d_in[10];
  const float* convv = (const float*)d_in[11];
  const float* rmsw  = (const float*)d_in[12];
  const float* Wout  = (const float*)d_in[13];
  const float* boutp = (const float*)d_in[14];
  float* out = (float*)d_out;

  float* ws = (float*)d_ws;
  const size_t SLOT = (size_t)Mrows * Dm;         // 8.39M floats = 32 MB
  float* kpre = ws + 0 * SLOT;                    // -> reused as Kp
  float* qpre = ws + 1 * SLOT;                    // -> reused as Qp
  float* vpre = ws + 2 * SLOT;                    // -> reused as Vp
  float* kcv  = ws + 3 * SLOT;                    // -> reused as Ograw
  float* qcv  = ws + 4 * SLOT;                    // -> reused as On
  float* vcv  = ws + 5 * SLOT;
  float* Bp   = ws + 6 * SLOT;
  float* beta = ws + 7 * SLOT;                    // 131072 floats
  float* Kp = kpre; float* Qp = qpre; float* Vp = vpre;
  float* Ograw = kcv; float* On = qcv;

  const dim3 gg(Dm / 64, Mrows / 64);             // (16, 128)
  gemm_xwt_kernel<<<gg, 512, 0, stream>>>(x, Wk, bk, kpre, Mrows, Dm, Dm);
  gemm_xwt_kernel<<<gg, 512, 0, stream>>>(x, Wq, bq, qpre, Mrows, Dm, Dm);
  gemm_xwt_kernel<<<gg, 512, 0, stream>>>(x, Wv, bv, vpre, Mrows, Dm, Dm);
  beta_kernel<<<(Mrows * H) / 256, 256, 0, stream>>>(x, Wbeta, bbeta, beta);

  const int convBlocks = (Mrows * Dm) / 256;      // 32768
  conv_kernel<<<convBlocks, 256, 0, stream>>>(kpre, convk, kcv);
  conv_kernel<<<convBlocks, 256, 0, stream>>>(qpre, convq, qcv);
  conv_kernel<<<convBlocks, 256, 0, stream>>>(vpre, convv, vcv);

  prep_kernel<<<(Mrows * H) / 8, 256, 0, stream>>>(kcv, qcv, vcv, beta, Kp, Qp, Vp, Bp);

  const size_t dsm = (size_t)9 * SZ * sizeof(float);   // ~150 KB LDS (<= 320 KB/WGP)
  delta_kernel<<<Bb * H, 512, dsm, stream>>>(Kp, Qp, Vp, Bp, Ograw);

  rms_kernel<<<(Bb * H * Lseq) / 8, 256, 0, stream>>>(Ograw, rmsw, On);

  gemm_xwt_kernel<<<gg, 512, 0, stream>>>(On, Wout, boutp, out, Mrows, Dm, Dm);
}